// Encoder_85246510891067
// MI455X (gfx1250) — compile-verified
//
#include <hip/hip_runtime.h>
#include <hip/hip_bf16.h>
#include <stdint.h>

// Problem constants (match reference).
#define DIMS    10000
#define LEVELS  256
#define SIZE    64
#define BATCH   32
#define TILE_D  128                       // d-columns per workgroup tile
#define NT      ((DIMS + TILE_D - 1) / TILE_D)   // 79 tiles

typedef __attribute__((ext_vector_type(4))) float f4;
typedef __attribute__((ext_vector_type(2))) float v2f;
typedef __attribute__((ext_vector_type(8))) float v8f;

// ---- CDNA5 async global->LDS (ASYNCcnt path) --------------------------------
__device__ __forceinline__ void async_load_b128(uint32_t lds_byte, const void* gaddr) {
  unsigned long long g = (unsigned long long)(uintptr_t)gaddr;
  asm volatile("global_load_async_to_lds_b128 %0, %1, off"
               :: "v"(lds_byte), "v"(g) : "memory");
}
__device__ __forceinline__ void wait_async0() {
  asm volatile("s_wait_asynccnt 0" ::: "memory");
}

// ---- Encode kernel ----------------------------------------------------------
// grid = (NT, YG); block = 256 (8 waves). Wave w handles batches 4w..4w+3.
// Lane l handles d = d0 + 4l .. 4l+3 (float4 per lane => ds_load_b128,
// conflict-free). LDS: vt[256][128] (128KB) + xw[64][128] (32KB) + yw[nrows][128].
extern "C" __global__ void __launch_bounds__(256)
hdc_encode(const float* __restrict__ vt, const float* __restrict__ xt,
           const float* __restrict__ yt, const int* __restrict__ img,
           float* __restrict__ part, int nrows)
{
  extern __shared__ float smem[];
  float* vt_s = smem;                              // LEVELS*TILE_D
  float* xw_s = smem + LEVELS * TILE_D;            // SIZE*TILE_D
  float* yw_s = xw_s + SIZE * TILE_D;              // nrows*TILE_D

  const int tid  = threadIdx.x;
  const int lane = tid & 31;
  const int wv   = __builtin_amdgcn_readfirstlane(tid >> 5);   // wave id (SGPR)
  const int d0   = blockIdx.x * TILE_D;
  const int y0   = blockIdx.y * nrows;
  const bool full = (d0 + TILE_D) <= DIMS;
  const uint32_t lds0 = __builtin_amdgcn_groupstaticsize();    // dyn-LDS base

  if (full) {
    // value_table tile: 256*128 f32 = 8192 float4 chunks (32 per thread)
    for (int f = tid; f < LEVELS * (TILE_D / 4); f += 256) {
      int lv = f >> 5, c = f & 31;
      async_load_b128(lds0 + (uint32_t)(lv * TILE_D + c * 4) * 4u,
                      vt + (size_t)lv * DIMS + d0 + c * 4);
    }
    // x_table tile: 64*128 f32 = 2048 float4
    for (int f = tid; f < SIZE * (TILE_D / 4); f += 256) {
      int r = f >> 5, c = f & 31;
      async_load_b128(lds0 + (uint32_t)((LEVELS + r) * TILE_D + c * 4) * 4u,
                      xt + (size_t)r * DIMS + d0 + c * 4);
    }
    // y_table tile for this y-group: nrows*128 f32
    for (int f = tid; f < nrows * (TILE_D / 4); f += 256) {
      int r = f >> 5, c = f & 31;
      async_load_b128(lds0 + (uint32_t)((LEVELS + SIZE + r) * TILE_D + c * 4) * 4u,
                      yt + (size_t)(y0 + r) * DIMS + d0 + c * 4);
    }
    wait_async0();
  } else {
    // Ragged last d-tile: guarded scalar fill, zero-padded.
    for (int f = tid; f < LEVELS * TILE_D; f += 256) {
      int lv = f >> 7, c = f & (TILE_D - 1), d = d0 + c;
      vt_s[f] = (d < DIMS) ? vt[(size_t)lv * DIMS + d] : 0.f;
    }
    for (int f = tid; f < SIZE * TILE_D; f += 256) {
      int r = f >> 7, c = f & (TILE_D - 1), d = d0 + c;
      xw_s[f] = (d < DIMS) ? xt[(size_t)r * DIMS + d] : 0.f;
    }
    for (int f = tid; f < nrows * TILE_D; f += 256) {
      int r = f >> 7, c = f & (TILE_D - 1), d = d0 + c;
      yw_s[f] = (d < DIMS) ? yt[(size_t)(y0 + r) * DIMS + d] : 0.f;
    }
  }
  __syncthreads();

  const f4* vt4 = (const f4*)vt_s;
  const f4* xw4 = (const f4*)xw_s;
  const f4* yw4 = (const f4*)yw_s;

  const int b0 = wv * 4;
  const int* __restrict__ im0 = img + (size_t)(b0 + 0) * SIZE * SIZE;
  const int* __restrict__ im1 = img + (size_t)(b0 + 1) * SIZE * SIZE;
  const int* __restrict__ im2 = img + (size_t)(b0 + 2) * SIZE * SIZE;
  const int* __restrict__ im3 = img + (size_t)(b0 + 3) * SIZE * SIZE;

  f4 acc0 = {0.f, 0.f, 0.f, 0.f}, acc1 = acc0, acc2 = acc0, acc3 = acc0;

  for (int r = 0; r < nrows; ++r) {
    const int y = y0 + r;
    const int* p0 = im0 + y * SIZE;
    const int* p1 = im1 + y * SIZE;
    const int* p2 = im2 + y * SIZE;
    const int* p3 = im3 + y * SIZE;
    f4 row0 = {0.f, 0.f, 0.f, 0.f}, row1 = row0, row2 = row0, row3 = row0;
#pragma unroll 8
    for (int x = 0; x < SIZE; ++x) {
      f4 xw = xw4[x * 32 + lane];                 // ds_load_b128, shared by 4 batches
      int i0 = p0[x], i1 = p1[x], i2 = p2[x], i3 = p3[x];  // uniform per wave
      row0 += vt4[i0 * 32 + lane] * xw;           // 4x ds_load_b128 gathers
      row1 += vt4[i1 * 32 + lane] * xw;           // + 16 v_fmac_f32
      row2 += vt4[i2 * 32 + lane] * xw;
      row3 += vt4[i3 * 32 + lane] * xw;
    }
    f4 yw = yw4[r * 32 + lane];                   // y-binding hoisted out of x loop
    acc0 += row0 * yw;
    acc1 += row1 * yw;
    acc2 += row2 * yw;
    acc3 += row3 * yw;
  }

  // Deterministic partial store: part layout [YG][BATCH][DIMS]
  float* wp = part + (size_t)blockIdx.y * BATCH * DIMS;
  const int dbase = d0 + lane * 4;
  f4 accs[4] = {acc0, acc1, acc2, acc3};
#pragma unroll
  for (int b = 0; b < 4; ++b) {
    float* dst = wp + (size_t)(b0 + b) * DIMS;
    if (full) {
      *(f4*)(dst + dbase) = accs[b];
    } else {
#pragma unroll
      for (int k = 0; k < 4; ++k)
        if (dbase + k < DIMS) dst[dbase + k] = accs[b][k];
    }
  }
}

// ---- WMMA reduction + hard quantize (nparts == 8 path) ----------------------
// out[i] = sign(sum_{g=0..7} part[g][i]) as a ones-GEMM:
//   C(16x16) += 1(16x4) x B(4x16), chained twice (8 K-slots = 8 partials).
// A is all-ones, so C[m,n] = sum_k B[k,n]: any bijection of the 8 partials
// onto the 8 (wmma#, B-VGPR, lane-half) K-slots yields the correct sum; only
// the documented N = lane[3:0] mapping of B and C is relied upon.
// 16 outputs per wave; 320000 outputs = 2500 blocks x 8 waves, no remainder.
extern "C" __global__ void __launch_bounds__(256)
hdc_reduce_wmma(const float* __restrict__ part, float* __restrict__ out)
{
  const int tid  = threadIdx.x;
  const int lane = tid & 31;
  const int wv   = tid >> 5;
  const int base = (blockIdx.x * 8 + wv) * 16;
  const int col  = lane & 15;
  const int half = lane >> 4;                   // selects K-slot pair per lane-half
  const size_t N = (size_t)BATCH * DIMS;
  const size_t i = (size_t)base + col;

  v2f a = {1.f, 1.f};
  v8f c = {0.f, 0.f, 0.f, 0.f, 0.f, 0.f, 0.f, 0.f};
  v2f b;

  // Partials 0..3 -> K-slots of WMMA #1
  b.x = part[(size_t)(0 + 2 * half) * N + i];
  b.y = part[(size_t)(1 + 2 * half) * N + i];
  c = __builtin_amdgcn_wmma_f32_16x16x4_f32(false, a, false, b, (short)0, c,
                                            false, false);
  // Partials 4..7 -> K-slots of WMMA #2 (C chained)
  b.x = part[(size_t)(4 + 2 * half) * N + i];
  b.y = part[(size_t)(5 + 2 * half) * N + i];
  c = __builtin_amdgcn_wmma_f32_16x16x4_f32(false, a, false, b, (short)0, c,
                                            false, false);

  if (lane < 16) out[i] = (c[0] > 0.f) ? 1.f : -1.f;  // hard_quantize: 0 -> -1
}

// ---- Scalar reduction fallback (nparts != 8) --------------------------------
extern "C" __global__ void __launch_bounds__(256)
hdc_reduce(const float* __restrict__ part, float* __restrict__ out, int nparts)
{
  int i = blockIdx.x * 256 + threadIdx.x;
  if (i < BATCH * DIMS) {
    float s = 0.f;
    for (int g = 0; g < nparts; ++g)
      s += part[(size_t)g * BATCH * DIMS + i];
    out[i] = (s > 0.f) ? 1.f : -1.f;
  }
}

// ---- Host launch ------------------------------------------------------------
extern "C" void kernel_launch(void* const* d_in, const int* in_sizes, int n_in,
                              void* d_out, int out_size, void* d_ws, size_t ws_size,
                              hipStream_t stream) {
  const float* vt  = (const float*)d_in[0];   // value_table [256,10000]
  const float* xt  = (const float*)d_in[1];   // x_table     [64,10000]
  const float* yt  = (const float*)d_in[2];   // y_table     [64,10000]
  const int*   img = (const int*)d_in[3];     // image       [32,1,64,64] int32
  float* out = (float*)d_out;                 // [32,10000] f32
  float* ws  = (float*)d_ws;

  // Split the 64 rows into YG y-groups for occupancy; shrink if ws is small.
  int YG = 8;
  while (YG > 1 && (size_t)YG * BATCH * DIMS * sizeof(float) > ws_size) YG >>= 1;
  float* pbuf = ws;
  if ((size_t)YG * BATCH * DIMS * sizeof(float) > ws_size) { pbuf = out; YG = 1; }
  const int nrows = SIZE / YG;

  const size_t shmem =
      ((size_t)LEVELS * TILE_D + (size_t)SIZE * TILE_D + (size_t)nrows * TILE_D) *
      sizeof(float);

  dim3 grid(NT, YG);
  hipLaunchKernelGGL(hdc_encode, grid, dim3(256), shmem, stream,
                     vt, xt, yt, img, pbuf, nrows);

  const int n = BATCH * DIMS;
  if (YG == 8) {
    // 16 outputs per wave, 8 waves per block -> n / 128 = 2500 blocks exactly.
    hipLaunchKernelGGL(hdc_reduce_wmma, dim3(n / 128), dim3(256), 0, stream,
                       pbuf, out);
  } else {
    hipLaunchKernelGGL(hdc_reduce, dim3((n + 255) / 256), dim3(256), 0, stream,
                       pbuf, out, YG);
  }
}